// PoseCNN_23871428231215
// MI455X (gfx1250) — compile-verified
//
#include <hip/hip_runtime.h>

// ---------------------------------------------------------------------------
// PoseCNN inference on MI455X (gfx1250, wave32).
//
// All conv layers (3x3 VGG convs + all 1x1 head convs) run as implicit-GEMM
// through V_WMMA_F32_16X16X4_F32 (fp32 in / fp32 accum, no precision loss —
// the workload is HBM/L2-bound at 23.3 TB/s, so the fp32 matrix pipe is
// nowhere near the bottleneck and bf16 down-conversion would only add error).
//
// Conv kernel: one wave computes a 16(Cout) x 64(pixels) tile with 4
// accumulators, so each A (weight) fragment load feeds 4 WMMAs. K-reduction
// is tap-major (r,s outer, ci inner): inner loop is pure pointer-stride
// arithmetic — no integer division — 2 A-loads + 8 B-loads + 4 WMMAs per
// 4-wide K step, unrolled x2.
//
// Deconvs run as gather-form VALU kernels (weights+inputs are L2-resident).
//
// Input order assumption (JAX pytree flatten, dict keys sorted):
//   [0]  x (1,3,384,512)
//   mask: [1]sc4_w [2]sc4_b [3]sc5_w [4]sc5_b [5]score_w [6]score_b
//         [7]up_w  [8]up_b  [9]up5_w [10]up5_b
//   vert: [11]sc4_w [12]sc4_b [13]sc5_w [14]sc5_b [15]up_w [16]up_b
//         [17]up5_w [18]up5_b [19]vp_w [20]vp_b
//   vgg : [21..46] = 13 x (w,b) in block order
//
// d_out = [score 22*196608][label 196608 (as float)][vertex 66*196608]
// d_ws needs ~218 MB (2 ping-pong bufs of 128*196608 floats + c4/c5 + slots).
// ---------------------------------------------------------------------------

typedef float v2f __attribute__((ext_vector_type(2)));
typedef float v8f __attribute__((ext_vector_type(8)));

// Implicit-GEMM conv: out[Cout, H*W] = W[Cout, Cin*KH*KW] x im2col(in) + bias.
// fp32 WMMA 16x16x4.  Fragment layouts (ISA 7.12.2):
//   A 16x4: lanes 0-15 -> M=lane, regs = K {k,k+1}; lanes 16-31 -> K {k+2,k+3}
//   B 4x16 mirrored: same K per half-wave, N = lane&15
//   C/D:    reg r -> M = r + 8*(lane>=16), N = lane&15
__global__ __launch_bounds__(32) void conv_wmma_kernel(
    const float* __restrict__ in, const float* __restrict__ wgt,
    const float* __restrict__ bias, float* __restrict__ out,
    int Cin, int Cout, int H, int W, int KH, int KW, int pad, int relu)
{
    const int lane = threadIdx.x;
    const int half = lane >> 4;   // 0: K pair {k,k+1}, 1: {k+2,k+3}
    const int lm   = lane & 15;
    const int npix = H * W;
    const int KHW  = KH * KW;
    const int Ktot = Cin * KHW;

    const int nBase = blockIdx.x * 64;   // 4 N-tiles of 16 pixels
    const int mBase = blockIdx.y * 16;

    // A row (clamped: rows >= Cout compute garbage we never store)
    const int m  = mBase + lm;
    const int mc = (m < Cout) ? m : (Cout - 1);
    const float* __restrict__ wrow = wgt + (size_t)mc * Ktot;

    // 4 B columns = 4 output pixels per lane (clamped; masked on store)
    int  p[4], oy[4], ox[4];
    bool pv[4];
    #pragma unroll
    for (int j = 0; j < 4; ++j) {
        p[j]  = nBase + j * 16 + lm;
        pv[j] = (p[j] < npix);
        const int pc = pv[j] ? p[j] : (npix - 1);
        oy[j] = pc / W;
        ox[j] = pc - oy[j] * W;
    }

    v8f acc0 = {}, acc1 = {}, acc2 = {}, acc3 = {};

    for (int r = 0; r < KH; ++r) {
        for (int s = 0; s < KW; ++s) {
            const float* __restrict__ ap = wrow + (r * KW + s);  // + ci*KHW
            const float* bp[4];
            bool bv[4];
            #pragma unroll
            for (int j = 0; j < 4; ++j) {
                const int iy  = oy[j] + r - pad;
                const int ix  = ox[j] + s - pad;
                const int iyc = (iy < 0) ? 0 : ((iy >= H) ? (H - 1) : iy);
                const int ixc = (ix < 0) ? 0 : ((ix >= W) ? (W - 1) : ix);
                bv[j] = pv[j] && ((unsigned)iy < (unsigned)H) &&
                                 ((unsigned)ix < (unsigned)W);
                bp[j] = in + (size_t)iyc * W + ixc;              // + ci*npix
            }
            #pragma unroll 2
            for (int ci = 0; ci < Cin; ci += 4) {
                const int  c0 = ci + half * 2;
                const int  c1 = c0 + 1;
                const bool v0 = (c0 < Cin);
                const bool v1 = (c1 < Cin);
                const int  c0c = v0 ? c0 : (Cin - 1);
                const int  c1c = v1 ? c1 : (Cin - 1);

                const float a0 = ap[(size_t)c0c * KHW];
                const float a1 = ap[(size_t)c1c * KHW];
                v2f a;
                a.x = v0 ? a0 : 0.0f;
                a.y = v1 ? a1 : 0.0f;

                v2f b;
                float t0, t1;

                t0 = bp[0][(size_t)c0c * npix];
                t1 = bp[0][(size_t)c1c * npix];
                b.x = (bv[0] && v0) ? t0 : 0.0f;
                b.y = (bv[0] && v1) ? t1 : 0.0f;
                acc0 = __builtin_amdgcn_wmma_f32_16x16x4_f32(
                           false, a, false, b, (short)0, acc0, false, false);

                t0 = bp[1][(size_t)c0c * npix];
                t1 = bp[1][(size_t)c1c * npix];
                b.x = (bv[1] && v0) ? t0 : 0.0f;
                b.y = (bv[1] && v1) ? t1 : 0.0f;
                acc1 = __builtin_amdgcn_wmma_f32_16x16x4_f32(
                           false, a, false, b, (short)0, acc1, false, false);

                t0 = bp[2][(size_t)c0c * npix];
                t1 = bp[2][(size_t)c1c * npix];
                b.x = (bv[2] && v0) ? t0 : 0.0f;
                b.y = (bv[2] && v1) ? t1 : 0.0f;
                acc2 = __builtin_amdgcn_wmma_f32_16x16x4_f32(
                           false, a, false, b, (short)0, acc2, false, false);

                t0 = bp[3][(size_t)c0c * npix];
                t1 = bp[3][(size_t)c1c * npix];
                b.x = (bv[3] && v0) ? t0 : 0.0f;
                b.y = (bv[3] && v1) ? t1 : 0.0f;
                acc3 = __builtin_amdgcn_wmma_f32_16x16x4_f32(
                           false, a, false, b, (short)0, acc3, false, false);
            }
        }
    }

    // Store D tiles (+bias, optional ReLU)
    v8f accs[4] = { acc0, acc1, acc2, acc3 };
    #pragma unroll
    for (int j = 0; j < 4; ++j) {
        if (!pv[j]) continue;
        #pragma unroll
        for (int rr = 0; rr < 8; ++rr) {
            const int mo = mBase + rr + half * 8;
            if (mo < Cout) {
                float v = accs[j][rr] + bias[mo];
                if (relu) v = fmaxf(v, 0.0f);
                out[(size_t)mo * npix + p[j]] = v;
            }
        }
    }
}

// ConvTranspose2d gather form: out[co,oy,ox] = b[co] +
//   sum_{ky,kx valid} sum_ci in[ci,iy,ix] * w[ci,co,ky,kx],
//   iy = (oy+pad-ky)/stride (integral), ix likewise.
__global__ void deconv_kernel(const float* __restrict__ in,
                              const float* __restrict__ wgt,
                              const float* __restrict__ bias,
                              float* __restrict__ out,
                              int Cin, int Cout, int Hin, int Win,
                              int K, int stride, int pad, int Hout, int Wout)
{
    const int idx   = blockIdx.x * blockDim.x + threadIdx.x;
    const int total = Cout * Hout * Wout;
    if (idx >= total) return;
    const int co  = idx / (Hout * Wout);
    const int rem = idx - co * Hout * Wout;
    const int oy  = rem / Wout;
    const int ox  = rem - oy * Wout;

    const int inHW = Hin * Win;
    const int KK   = K * K;
    const int wci  = Cout * KK;   // weight stride per ci
    float acc = bias[co];

    const int ky0 = (oy + pad) % stride;
    const int kx0 = (ox + pad) % stride;
    for (int ky = ky0; ky < K; ky += stride) {
        const int ty = oy + pad - ky;
        if (ty < 0) break;
        const int iy = ty / stride;
        if (iy >= Hin) continue;
        for (int kx = kx0; kx < K; kx += stride) {
            const int tx = ox + pad - kx;
            if (tx < 0) break;
            const int ix = tx / stride;
            if (ix >= Win) continue;
            const float* __restrict__ ip = in + iy * Win + ix;
            const float* __restrict__ wp = wgt + co * KK + ky * K + kx;
            for (int ci = 0; ci < Cin; ++ci)
                acc = fmaf(ip[(size_t)ci * inHW], wp[(size_t)ci * wci], acc);
        }
    }
    out[idx] = acc;
}

__global__ void maxpool2_kernel(const float* __restrict__ in,
                                float* __restrict__ out, int C, int H, int W)
{
    const int Ho = H >> 1, Wo = W >> 1;
    const int idx   = blockIdx.x * blockDim.x + threadIdx.x;
    const int total = C * Ho * Wo;
    if (idx >= total) return;
    const int c   = idx / (Ho * Wo);
    const int rem = idx - c * Ho * Wo;
    const int oy  = rem / Wo;
    const int ox  = rem - oy * Wo;
    const float* __restrict__ ip = in + ((size_t)c * H + oy * 2) * W + ox * 2;
    out[idx] = fmaxf(fmaxf(ip[0], ip[1]), fmaxf(ip[W], ip[W + 1]));
}

__global__ void add_kernel(const float* __restrict__ a,
                           const float* __restrict__ b,
                           float* __restrict__ out, int n)
{
    const int i = blockIdx.x * blockDim.x + threadIdx.x;
    if (i < n) out[i] = a[i] + b[i];
}

// argmax over channel axis of score (softmax is monotone -> argmax(score)).
__global__ void argmax_kernel(const float* __restrict__ score,
                              float* __restrict__ label, int C, int npix)
{
    const int p = blockIdx.x * blockDim.x + threadIdx.x;
    if (p >= npix) return;
    float best = score[p];
    int   bi   = 0;
    for (int c = 1; c < C; ++c) {
        const float v = score[(size_t)c * npix + p];
        if (v > best) { best = v; bi = c; }
    }
    label[p] = (float)bi;
}

extern "C" void kernel_launch(void* const* d_in, const int* in_sizes, int n_in,
                              void* d_out, int out_size, void* d_ws, size_t ws_size,
                              hipStream_t stream)
{
    (void)in_sizes; (void)n_in; (void)out_size; (void)ws_size;

    const float* X = (const float*)d_in[0];
    // mask head params
    const float* m_sc4_w   = (const float*)d_in[1];
    const float* m_sc4_b   = (const float*)d_in[2];
    const float* m_sc5_w   = (const float*)d_in[3];
    const float* m_sc5_b   = (const float*)d_in[4];
    const float* m_score_w = (const float*)d_in[5];
    const float* m_score_b = (const float*)d_in[6];
    const float* m_up_w    = (const float*)d_in[7];
    const float* m_up_b    = (const float*)d_in[8];
    const float* m_up5_w   = (const float*)d_in[9];
    const float* m_up5_b   = (const float*)d_in[10];
    // vertex head params
    const float* v_sc4_w   = (const float*)d_in[11];
    const float* v_sc4_b   = (const float*)d_in[12];
    const float* v_sc5_w   = (const float*)d_in[13];
    const float* v_sc5_b   = (const float*)d_in[14];
    const float* v_up_w    = (const float*)d_in[15];
    const float* v_up_b    = (const float*)d_in[16];
    const float* v_up5_w   = (const float*)d_in[17];
    const float* v_up5_b   = (const float*)d_in[18];
    const float* v_vp_w    = (const float*)d_in[19];
    const float* v_vp_b    = (const float*)d_in[20];
    // vgg params
    const float* vw[13]; const float* vb[13];
    for (int i = 0; i < 13; ++i) {
        vw[i] = (const float*)d_in[21 + 2 * i];
        vb[i] = (const float*)d_in[22 + 2 * i];
    }

    const int FULL = 384 * 512;               // 196608 pixels at full res
    float* ws   = (float*)d_ws;
    float* bufA = ws;                          // 128*FULL floats
    float* bufB = ws + (size_t)128 * FULL;     // 128*FULL floats
    float* c4   = ws + (size_t)256 * FULL;     // 512*48*64  = 8*FULL
    float* c5   = ws + (size_t)264 * FULL;     // 512*24*32  = 2*FULL
    float* s4   = ws + (size_t)266 * FULL;
    float* s5   = ws + (size_t)267 * FULL;
    float* up5m = ws + (size_t)268 * FULL;
    float* summ = ws + (size_t)269 * FULL;
    float* v4   = ws + (size_t)270 * FULL;     // 2*FULL
    float* vup5 = ws + (size_t)272 * FULL;     // 2*FULL
    float* sumv = ws + (size_t)274 * FULL;     // 2*FULL
    float* v5   = ws + (size_t)276 * FULL;

    float* outScore  = (float*)d_out;
    float* outLabel  = outScore + (size_t)22 * FULL;
    float* outVertex = outScore + (size_t)23 * FULL;

    auto conv = [&](const float* in, const float* w, const float* b, float* o,
                    int Cin, int Cout, int H, int W, int K, int pad, int relu) {
        dim3 g((H * W + 63) / 64, (Cout + 15) / 16);
        conv_wmma_kernel<<<g, 32, 0, stream>>>(in, w, b, o, Cin, Cout, H, W, K, K, pad, relu);
    };
    auto pool = [&](const float* in, float* o, int C, int H, int W) {
        int tot = C * (H / 2) * (W / 2);
        maxpool2_kernel<<<(tot + 255) / 256, 256, 0, stream>>>(in, o, C, H, W);
    };
    auto deconv = [&](const float* in, const float* w, const float* b, float* o,
                      int Cin, int Cout, int Hin, int Win, int K, int s, int pad) {
        int Hout = Hin * s, Wout = Win * s;
        int tot = Cout * Hout * Wout;
        deconv_kernel<<<(tot + 255) / 256, 256, 0, stream>>>(
            in, w, b, o, Cin, Cout, Hin, Win, K, s, pad, Hout, Wout);
    };
    auto addv = [&](const float* a, const float* b, float* o, int n) {
        add_kernel<<<(n + 255) / 256, 256, 0, stream>>>(a, b, o, n);
    };

    // ---------------- VGG16 backbone ----------------
    conv(X,    vw[0],  vb[0],  bufA, 3,   64,  384, 512, 3, 1, 1);
    conv(bufA, vw[1],  vb[1],  bufB, 64,  64,  384, 512, 3, 1, 1);
    pool(bufB, bufA, 64, 384, 512);
    conv(bufA, vw[2],  vb[2],  bufB, 64,  128, 192, 256, 3, 1, 1);
    conv(bufB, vw[3],  vb[3],  bufA, 128, 128, 192, 256, 3, 1, 1);
    pool(bufA, bufB, 128, 192, 256);
    conv(bufB, vw[4],  vb[4],  bufA, 128, 256, 96, 128, 3, 1, 1);
    conv(bufA, vw[5],  vb[5],  bufB, 256, 256, 96, 128, 3, 1, 1);
    conv(bufB, vw[6],  vb[6],  bufA, 256, 256, 96, 128, 3, 1, 1);
    pool(bufA, bufB, 256, 96, 128);
    conv(bufB, vw[7],  vb[7],  bufA, 256, 512, 48, 64, 3, 1, 1);
    conv(bufA, vw[8],  vb[8],  bufB, 512, 512, 48, 64, 3, 1, 1);
    conv(bufB, vw[9],  vb[9],  c4,   512, 512, 48, 64, 3, 1, 1);   // c4 (/8)
    pool(c4,   bufA, 512, 48, 64);
    conv(bufA, vw[10], vb[10], bufB, 512, 512, 24, 32, 3, 1, 1);
    conv(bufB, vw[11], vb[11], bufA, 512, 512, 24, 32, 3, 1, 1);
    conv(bufA, vw[12], vb[12], c5,   512, 512, 24, 32, 3, 1, 1);   // c5 (/16)

    // ---------------- Semantic mask head ----------------
    conv(c4, m_sc4_w, m_sc4_b, s4, 512, 64, 48, 64, 1, 0, 1);
    conv(c5, m_sc5_w, m_sc5_b, s5, 512, 64, 24, 32, 1, 0, 1);
    deconv(s5, m_up5_w, m_up5_b, up5m, 64, 64, 24, 32, 4, 2, 1);   // /16 -> /8
    addv(s4, up5m, summ, 64 * 48 * 64);
    deconv(summ, m_up_w, m_up_b, bufA, 64, 64, 48, 64, 16, 8, 4);  // /8 -> full
    conv(bufA, m_score_w, m_score_b, outScore, 64, 22, 384, 512, 1, 0, 1);
    argmax_kernel<<<(FULL + 255) / 256, 256, 0, stream>>>(outScore, outLabel, 22, FULL);

    // ---------------- Vertex regression head ----------------
    conv(c4, v_sc4_w, v_sc4_b, v4, 512, 128, 48, 64, 1, 0, 0);
    conv(c5, v_sc5_w, v_sc5_b, v5, 512, 128, 24, 32, 1, 0, 0);
    deconv(v5, v_up5_w, v_up5_b, vup5, 128, 128, 24, 32, 4, 2, 1);
    addv(v4, vup5, sumv, 128 * 48 * 64);
    deconv(sumv, v_up_w, v_up_b, bufB, 128, 128, 48, 64, 16, 8, 4);
    conv(bufB, v_vp_w, v_vp_b, outVertex, 128, 66, 384, 512, 1, 0, 0);
}